// MasifSiteNet_6940667150380
// MI455X (gfx1250) — compile-verified
//
#include <hip/hip_runtime.h>

#define N_SURF  40000
#define N_GRAPH 5000
#define DMODEL  128
#define NBATCH  16
#define BN_EPS  1e-5f
#define SLOPE   0.2f

// fused-kernel geometry
#define ROWS_PER_BLOCK 160                 // 10 M-tiles of 16; 250 blocks exactly
#define NTILES_PER_BLOCK (ROWS_PER_BLOCK / 16)
#define WPITCH  144                        // float2 pitch per packed-W1 row: 288 dwords ≡ 32 (mod 64)
#define XPITCH  132                        // dword pitch for x/h tiles (bank-conflict-free)

typedef __attribute__((ext_vector_type(2))) float v2f;
typedef __attribute__((ext_vector_type(8))) float v8f;

// ---------------------------------------------------------------------------
// Kernel A: batch ranges. graph_batch is sorted; ranges[b] = lower_bound(b).
// ---------------------------------------------------------------------------
__global__ void batch_ranges_kernel(const int* __restrict__ gbatch,
                                    int* __restrict__ ranges) {
  int t = threadIdx.x;
  if (t > NBATCH) return;
  int lo = 0, hi = N_GRAPH;
  while (lo < hi) {
    int mid = (lo + hi) >> 1;
    if (gbatch[mid] < t) lo = mid + 1; else hi = mid;
  }
  ranges[t] = lo;
}

// ---------------------------------------------------------------------------
// Kernel B: masked nearest neighbor, restricted to the point's own batch
// range (both batch arrays are sorted). graph_pos (60KB) is L2-resident.
// Strict '<' + ascending scan == jnp.argmin first-match semantics.
// ---------------------------------------------------------------------------
__global__ __launch_bounds__(256) void nn_kernel(
    const float* __restrict__ spos, const int* __restrict__ sbatch,
    const float* __restrict__ gpos, const int* __restrict__ ranges,
    int* __restrict__ nearest) {
  int i = blockIdx.x * blockDim.x + threadIdx.x;
  if (i >= N_SURF) return;
  float sx = spos[3 * i + 0];
  float sy = spos[3 * i + 1];
  float sz = spos[3 * i + 2];
  int b  = sbatch[i];
  int rs = ranges[b];
  int re = ranges[b + 1];
  float best = __builtin_inff();
  int bestj = 0;
  for (int j = rs; j < re; ++j) {
    float dx = sx - gpos[3 * j + 0];
    float dy = sy - gpos[3 * j + 1];
    float dz = sz - gpos[3 * j + 2];
    float d2 = fmaf(dx, dx, fmaf(dy, dy, dz * dz));
    if (d2 < best) { best = d2; bestj = j; }
  }
  nearest[i] = bestj;
}

// ---------------------------------------------------------------------------
// Kernel P: pack W1 into per-lane WMMA B-fragment order.
// Packed row r = kk*2 + half (64 rows, pitch WPITCH float2): entry n holds
// (W1[k0*128+n], W1[(k0+1)*128+n]) with k0 = 4*kk + 2*half, so each lane's
// B fragment is ONE contiguous b64 instead of two stride-512B b32 loads.
// ---------------------------------------------------------------------------
__global__ __launch_bounds__(256) void pack_w1_kernel(const float* __restrict__ W1,
                                                      float2* __restrict__ W1p) {
  int idx = blockIdx.x * blockDim.x + threadIdx.x;   // 0 .. 8191
  if (idx >= 64 * DMODEL) return;
  int r = idx >> 7;          // kk*2 + half
  int n = idx & 127;
  int k0 = (r >> 1) * 4 + (r & 1) * 2;
  float2 p;
  p.x = W1[k0 * DMODEL + n];
  p.y = W1[(k0 + 1) * DMODEL + n];
  W1p[r * WPITCH + n] = p;
}

// ---------------------------------------------------------------------------
// Kernel C: fused gather + (x@W1+b1) via V_WMMA_F32_16X16X4_F32 + BN + leaky
// + row-dot with W2 + leaky. Packed W1 is staged in LDS ONCE per block and
// amortized over NTILES_PER_BLOCK M-tiles (cuts W1 L2 traffic ~10x).
//
// A-fragment (32-bit 16x4, ISA 7.12.2): lane = M + 16*(K>=2), VGPR v -> K=2h+v.
// C/D (16x16 f32): VGPR r: lanes 0-15 -> M=r, lanes 16-31 -> M=r+8, N=lane%16.
// Dynamic LDS layout: [ W1p 64*WPITCH float2 | x 16*XPITCH f32 | h 16*XPITCH f32 ]
// ---------------------------------------------------------------------------
__global__ __launch_bounds__(256) void fused_gemm_kernel(
    const float*  __restrict__ gfeat,   // [N_GRAPH, D]
    const int*    __restrict__ nearest, // [N_SURF]
    const float2* __restrict__ W1p,     // packed W1, 64*WPITCH float2
    const float*  __restrict__ b1,
    const float*  __restrict__ gamma_v,
    const float*  __restrict__ beta_v,
    const float*  __restrict__ mean_v,
    const float*  __restrict__ var_v,
    const float*  __restrict__ W2,      // [D]
    const float*  __restrict__ b2,      // [1]
    float* __restrict__ out) {          // [N_SURF]
  extern __shared__ __align__(16) char smem[];
  float2* w_lds = reinterpret_cast<float2*>(smem);                       // 64*WPITCH
  float*  x_lds = reinterpret_cast<float*>(smem + 64 * WPITCH * 8);      // 16*XPITCH
  float*  h_lds = x_lds + 16 * XPITCH;                                   // 16*XPITCH

  const int tid  = threadIdx.x;
  const int wave = tid >> 5;             // 0..7 -> N-tile
  const int lane = tid & 31;
  const int half = lane >> 4;            // 0 or 1
  const int mrow = lane & 15;
  const int ncol = wave * 16 + (lane & 15);

  // --- stage packed W1 in LDS once (64KB payload) ---
  for (int i = tid; i < 64 * WPITCH; i += 256) w_lds[i] = W1p[i];

  // per-column epilogue constants (hoisted out of the tile loop)
  const float scale = gamma_v[ncol] * rsqrtf(var_v[ncol] + BN_EPS);
  const float shift = (b1[ncol] - mean_v[ncol]) * scale + beta_v[ncol];
  const float bias2 = b2[0];

  const int row = tid >> 4;        // gather helpers
  const int seg = tid & 15;

  for (int t = 0; t < NTILES_PER_BLOCK; ++t) {
    const int m0 = blockIdx.x * ROWS_PER_BLOCK + t * 16;

    __syncthreads();   // prev iter's h_lds readers + x_lds readers done; w_lds ready (t==0)

    // --- cooperative gather of 16 A rows (16 x 128 f32) ---
    {
      int nn = nearest[m0 + row];
      const float4* src = reinterpret_cast<const float4*>(gfeat + nn * DMODEL + seg * 8);
      float4 p0 = src[0];
      float4 p1 = src[1];
      float4* dst = reinterpret_cast<float4*>(&x_lds[row * XPITCH + seg * 8]);
      dst[0] = p0;
      dst[1] = p1;
    }
    __syncthreads();

    // --- 16x16 tile GEMM: 32 chained f32 WMMAs over K=128 ---
    v8f c = {0.f, 0.f, 0.f, 0.f, 0.f, 0.f, 0.f, 0.f};
#pragma unroll
    for (int kk = 0; kk < 32; ++kk) {
      const int kb = kk * 4 + half * 2;
      v2f a, bm;
      a.x = x_lds[mrow * XPITCH + kb];
      a.y = x_lds[mrow * XPITCH + kb + 1];
      float2 bp = w_lds[(kk * 2 + half) * WPITCH + ncol];
      bm.x = bp.x;
      bm.y = bp.y;
      c = __builtin_amdgcn_wmma_f32_16x16x4_f32(
          /*neg_a=*/false, a, /*neg_b=*/false, bm,
          /*c_mod=*/(short)0, c, /*reuse_a=*/false, /*reuse_b=*/false);
    }

    // --- epilogue: +b1, BatchNorm (running stats), leaky relu -> LDS ---
#pragma unroll
    for (int r = 0; r < 8; ++r) {
      float hv = fmaf(c[r], scale, shift);
      hv = (hv >= 0.f) ? hv : hv * SLOPE;
      h_lds[(r + 8 * half) * XPITCH + ncol] = hv;
    }
    __syncthreads();

    // --- fixed-order row dot with W2 (OUT=1), final leaky ---
    if (tid < 16) {
      float acc = 0.f;
#pragma unroll 8
      for (int k = 0; k < DMODEL; ++k)
        acc = fmaf(h_lds[tid * XPITCH + k], W2[k], acc);
      acc += bias2;
      out[m0 + tid] = (acc >= 0.f) ? acc : acc * SLOPE;
    }
  }
}

// ---------------------------------------------------------------------------
extern "C" void kernel_launch(void* const* d_in, const int* in_sizes, int n_in,
                              void* d_out, int out_size, void* d_ws, size_t ws_size,
                              hipStream_t stream) {
  const float* gfeat   = (const float*)d_in[0];
  const float* gpos    = (const float*)d_in[1];
  const float* spos    = (const float*)d_in[2];
  const int*   sbatch  = (const int*)d_in[3];
  const int*   gbatch  = (const int*)d_in[4];
  const float* W1      = (const float*)d_in[5];
  const float* b1      = (const float*)d_in[6];
  const float* gamma_v = (const float*)d_in[7];
  const float* beta_v  = (const float*)d_in[8];
  const float* mean_v  = (const float*)d_in[9];
  const float* var_v   = (const float*)d_in[10];
  const float* W2      = (const float*)d_in[11];
  const float* b2      = (const float*)d_in[12];
  float* out = (float*)d_out;

  // workspace layout
  int*    ranges  = (int*)d_ws;                                    // 17 ints
  int*    nearest = (int*)((char*)d_ws + 1024);                    // 40000 ints
  float2* W1p     = (float2*)((char*)d_ws + 1024 + 160768);        // 64*WPITCH float2

  batch_ranges_kernel<<<1, 32, 0, stream>>>(gbatch, ranges);
  pack_w1_kernel<<<(64 * DMODEL + 255) / 256, 256, 0, stream>>>(W1, W1p);
  nn_kernel<<<(N_SURF + 255) / 256, 256, 0, stream>>>(spos, sbatch, gpos, ranges, nearest);

  const size_t lds_bytes = 64 * WPITCH * sizeof(float2) + 2 * 16 * XPITCH * sizeof(float);
  fused_gemm_kernel<<<N_SURF / ROWS_PER_BLOCK, 256, lds_bytes, stream>>>(
      gfeat, nearest, W1p, b1, gamma_v, beta_v, mean_v, var_v, W2, b2, out);
}